// LieDissipative_34016140984980
// MI455X (gfx1250) — compile-verified
//
#include <hip/hip_runtime.h>

// LieDissipative: x_t = expm(A*dt)^t @ x0, A = S - S^T - diag(softplus(D)).
// DIM=128, BATCH=64, T=4096. Two-level blocked scan (B=64) to break the
// serial chain; f64 expm on VALU, f32 scan on V_WMMA_F32_16X16X4_F32.
// Round 3: all copy loops get compile-time trip counts (previous codegen
// produced exec-masked dynamic loops with serialized LDS round-trips);
// per-step emit vectorized (ds_load_b64 + NT global_store_b128).

#define DIM    128
#define BATCH  64
#define TSTEPS 4096
#define BLK    64                 // time-block length
#define NBLK   (TSTEPS / BLK)     // 64 blocks
#define NTH    1024               // 32 waves per workgroup (wave32)
#define EP     129                // LDS pitch (floats) for E/P: avoids stride-128 bank conflict
#define YP     76                 // LDS pitch (floats) for Y: disjoint half-wave bank spans
#define KSTEPS (DIM / 4)          // 32 WMMA k-steps of K=4

typedef float  v2f __attribute__((ext_vector_type(2)));
typedef float  v4f __attribute__((ext_vector_type(4)));
typedef float  v8f __attribute__((ext_vector_type(8)));
typedef double v2d __attribute__((ext_vector_type(2)));

// ---------------------------------------------------------------------------
// Stage A: build A*dt, E = expm(A*dt) (12-term Horner), P = E^64. All f64.
// Single workgroup; X and H live in dynamic LDS (2 * 128*128 * 8B = 256 KB).
// ---------------------------------------------------------------------------
__global__ __launch_bounds__(NTH, 1)
void lie_expm_f64(const double* __restrict__ t,
                  const double* __restrict__ S_upper,
                  const double* __restrict__ D_raw,
                  float* __restrict__ E32,
                  float* __restrict__ P32) {
  extern __shared__ char smem[];
  double* X = (double*)smem;       // A*dt
  double* H = X + DIM * DIM;       // Horner accumulator / result
  const int tid = threadIdx.x;
  const double dt = t[1] - t[0];

  // Build X = A*dt and H = I. triu_indices(k=1) row-major:
  // idx(i,j) = i*(2*DIM-i-1)/2 + (j-i-1)
#pragma unroll
  for (int q = 0; q < (DIM * DIM) / NTH; ++q) {
    const int e = q * NTH + tid;
    const int i = e >> 7, j = e & (DIM - 1);
    double v;
    if (i == j) {
      v = -log1p(exp(D_raw[i]));                         // -softplus(D)
    } else if (i < j) {
      v = S_upper[i * (2 * DIM - i - 1) / 2 + (j - i - 1)];
    } else {
      v = -S_upper[j * (2 * DIM - j - 1) / 2 + (i - j - 1)];
    }
    X[e] = v * dt;
    H[e] = (i == j) ? 1.0 : 0.0;
  }
  __syncthreads();

  const int col = tid & (DIM - 1);       // this thread's column
  const int r0  = (tid >> 7) << 4;       // this thread's 16-row group

  // exp(X) = I + X(I + X/2(I + ... (I + X/12)...)); ||X|| ~ 0.07 -> eps-level.
  for (int k = 12; k >= 1; --k) {
    double acc[16];
#pragma unroll
    for (int r = 0; r < 16; ++r) acc[r] = 0.0;
    for (int j = 0; j < DIM; ++j) {
      const double hv = H[j * DIM + col];          // bank-clean: lanes = cols
#pragma unroll
      for (int r = 0; r < 16; ++r)
        acc[r] = fma(X[(r0 + r) * DIM + j], hv, acc[r]);  // broadcast read
    }
    const double invk = 1.0 / (double)k;
    __syncthreads();                               // reads of H done
#pragma unroll
    for (int r = 0; r < 16; ++r) {
      double v = acc[r] * invk;
      if ((r0 + r) == col) v += 1.0;
      H[(r0 + r) * DIM + col] = v;
    }
    __syncthreads();
  }
#pragma unroll
  for (int q = 0; q < (DIM * DIM) / NTH; ++q) {
    const int e = q * NTH + tid;
    E32[e] = (float)H[e];
  }

  // P = E^64 via 6 squarings, ping-pong H <-> X (X no longer needed).
  double* src = H;
  double* dst = X;
  for (int s = 0; s < 6; ++s) {
    double acc[16];
#pragma unroll
    for (int r = 0; r < 16; ++r) acc[r] = 0.0;
    for (int j = 0; j < DIM; ++j) {
      const double hv = src[j * DIM + col];
#pragma unroll
      for (int r = 0; r < 16; ++r)
        acc[r] = fma(src[(r0 + r) * DIM + j], hv, acc[r]);
    }
#pragma unroll
    for (int r = 0; r < 16; ++r) dst[(r0 + r) * DIM + col] = acc[r];
    __syncthreads();
    double* tmp = src; src = dst; dst = tmp;
  }
#pragma unroll
  for (int q = 0; q < (DIM * DIM) / NTH; ++q) {
    const int e = q * NTH + tid;
    P32[e] = (float)src[e];
  }
}

// ---------------------------------------------------------------------------
// WMMA helpers (wave32, V_WMMA_F32_16X16X4_F32).
// 32 waves: wave w owns C tile (mt = w>>2, nt = w&3).
// ISA layouts: A frag lane L -> M=L&15, K pair 2h..2h+1 (h=L>>4);
// B frag lane L -> N=L&15, rows 2h..2h+1 of the K-slice;
// D VGPR r -> M = tile_m + r + 8h, N = L&15.
// ---------------------------------------------------------------------------

// Load this wave's 32 A-matrix K=4 fragments into registers (64 VGPRs).
// The A operand (E or P) is invariant across the whole step chain.
__device__ __forceinline__
void load_afrag(const float* __restrict__ Elds, int tid, v2f* __restrict__ af) {
  const int lane = tid & 31;
  const int mt   = (tid >> 5) >> 2;     // wave's M tile
  const int h    = lane >> 4;
  const int m    = (mt << 4) + (lane & 15);
  const float* arow = Elds + m * EP + 2 * h;
#pragma unroll
  for (int kk = 0; kk < KSTEPS; ++kk) {
    af[kk].x = arow[4 * kk];
    af[kk].y = arow[4 * kk + 1];
  }
}

// nxt[128x64] = A(regs) @ cur[128x64]; 32 accumulating WMMAs per wave.
__device__ __forceinline__
void wmma_step(const v2f* __restrict__ af,
               const float* __restrict__ cur,
               float* __restrict__ nxt,
               int tid) {
  const int lane = tid & 31;
  const int wave = tid >> 5;
  const int mt   = wave >> 2;
  const int nt   = wave & 3;
  const int h    = lane >> 4;
  const int n    = (nt << 4) + (lane & 15);

  v8f acc = {0.f, 0.f, 0.f, 0.f, 0.f, 0.f, 0.f, 0.f};
  const float* bcol = cur + (2 * h) * YP + n;
#pragma unroll
  for (int kk = 0; kk < KSTEPS; ++kk) {
    v2f b;
    b.x = bcol[(4 * kk) * YP];
    b.y = bcol[(4 * kk) * YP + YP];
    acc = __builtin_amdgcn_wmma_f32_16x16x4_f32(
        /*neg_a=*/false, af[kk], /*neg_b=*/false, b,
        /*c_mod=*/(short)0, acc, /*reuse_a=*/false, /*reuse_b=*/false);
  }
  float* drow = nxt + ((mt << 4) + (h << 3)) * YP + n;
#pragma unroll
  for (int r = 0; r < 8; ++r) drow[r * YP] = acc[r];
}

// ---------------------------------------------------------------------------
// Stage B: serial chain of block-start states Y_{k+1} = P @ Y_k (64 steps).
// Yb[k] = E^(64k) x0, written f32 to workspace (hot in L2 for stage C).
// ---------------------------------------------------------------------------
__global__ __launch_bounds__(NTH, 1)
void lie_blockstarts(const double* __restrict__ x0,
                     const float* __restrict__ P32,
                     float* __restrict__ Yb) {
  extern __shared__ char smem[];
  float* P  = (float*)smem;             // DIM*EP
  float* Y0 = P + DIM * EP;             // DIM*YP
  float* Y1 = Y0 + DIM * YP;            // DIM*YP
  const int tid = threadIdx.x;

  // Per-thread 8-element row segment (contiguous in both LDS and global).
  const int row = tid >> 3;             // 0..127
  const int c0  = (tid & 7) * 8;        // 0,8,...,56

#pragma unroll
  for (int q = 0; q < (DIM * DIM) / NTH; ++q) {  // 16 per thread, pipelined
    const int e = q * NTH + tid;
    P[(e >> 7) * EP + (e & (DIM - 1))] = P32[e];
  }
  {
    const double* gp = x0 + row * BATCH + c0;
    float* sp = Y0 + row * YP + c0;
#pragma unroll
    for (int q = 0; q < 8; ++q) sp[q] = (float)gp[q];
  }
  __syncthreads();

  v2f af[KSTEPS];
  load_afrag(P, tid, af);

  float* cur = Y0;
  float* nxt = Y1;
  for (int k = 0; k < NBLK; ++k) {
    {
      const float* sp = cur + row * YP + c0;    // 16B-aligned (YP*4 = 304)
      float* gp = Yb + k * DIM * BATCH + row * BATCH + c0;
      const v4f f0 = *(const v4f*)(sp);
      const v4f f1 = *(const v4f*)(sp + 4);
      *(v4f*)(gp)     = f0;
      *(v4f*)(gp + 4) = f1;
    }
    if (k < NBLK - 1) {                 // uniform branch: EXEC stays all-1s
      wmma_step(af, cur, nxt, tid);
      __syncthreads();
      float* tmp = cur; cur = nxt; nxt = tmp;
    }
  }
}

// ---------------------------------------------------------------------------
// Stage C: 64 blocks in parallel. Block b: start from Yb[b], emit 64 states
// (f32 -> f64, non-temporal b128 stores: 268 MB write-once stream > 192 MB
// L2), stepping Y <- E @ Y with WMMA.
// ---------------------------------------------------------------------------
__global__ __launch_bounds__(NTH, 1)
void lie_scan(const float* __restrict__ E32,
              const float* __restrict__ Yb,
              double* __restrict__ out) {
  extern __shared__ char smem[];
  float* E  = (float*)smem;             // DIM*EP
  float* Y0 = E + DIM * EP;
  float* Y1 = Y0 + DIM * YP;
  const int tid = threadIdx.x;
  const int b   = blockIdx.x;

  const int row = tid >> 3;             // 0..127
  const int c0  = (tid & 7) * 8;        // 0,8,...,56

#pragma unroll
  for (int q = 0; q < (DIM * DIM) / NTH; ++q) {
    const int e = q * NTH + tid;
    E[(e >> 7) * EP + (e & (DIM - 1))] = E32[e];
  }
  {
    const float* gp = Yb + b * DIM * BATCH + row * BATCH + c0;
    float* sp = Y0 + row * YP + c0;
    *(v4f*)(sp)     = *(const v4f*)(gp);
    *(v4f*)(sp + 4) = *(const v4f*)(gp + 4);
  }
  __syncthreads();

  v2f af[KSTEPS];
  load_afrag(E, tid, af);

  float* cur = Y0;
  float* nxt = Y1;
  const size_t base = (size_t)b * BLK * DIM * BATCH;
  for (int j = 0; j < BLK; ++j) {
    {
      const float* sp = cur + row * YP + c0;
      double* dp = out + base + (size_t)j * (DIM * BATCH) + row * BATCH + c0;
#pragma unroll
      for (int q = 0; q < 4; ++q) {
        const v2f f = *(const v2f*)(sp + 2 * q);      // ds_load_b64
        v2d d; d.x = (double)f.x; d.y = (double)f.y;
        __builtin_nontemporal_store(d, (v2d*)dp + q); // NT global_store_b128
      }
    }
    if (j < BLK - 1) {
      wmma_step(af, cur, nxt, tid);
      __syncthreads();
      float* tmp = cur; cur = nxt; nxt = tmp;
    }
  }
}

// ---------------------------------------------------------------------------
// Host launcher. Inputs (f64): t[4096], x0[128*64], S_upper[8128], D_raw[128].
// Output: f64 [4096,128,64]. Workspace: E32 | P32 | Yb (f32), ~2.2 MB.
// ---------------------------------------------------------------------------
extern "C" void kernel_launch(void* const* d_in, const int* in_sizes, int n_in,
                              void* d_out, int out_size, void* d_ws, size_t ws_size,
                              hipStream_t stream) {
  (void)in_sizes; (void)n_in; (void)out_size; (void)ws_size;
  const double* t  = (const double*)d_in[0];
  const double* x0 = (const double*)d_in[1];
  const double* Su = (const double*)d_in[2];
  const double* Dr = (const double*)d_in[3];
  double* out = (double*)d_out;

  float* wsf = (float*)d_ws;
  float* E32 = wsf;                       // 16384 floats
  float* P32 = wsf + DIM * DIM;           // 16384 floats
  float* Yb  = wsf + 2 * DIM * DIM;       // 64 * 8192 floats

  const size_t smemA  = (size_t)2 * DIM * DIM * sizeof(double);            // 256 KB
  const size_t smemBC = (size_t)(DIM * EP + 2 * DIM * YP) * sizeof(float); // ~140 KB

  lie_expm_f64<<<1, NTH, smemA, stream>>>(t, Su, Dr, E32, P32);
  lie_blockstarts<<<1, NTH, smemBC, stream>>>(x0, P32, Yb);
  lie_scan<<<NBLK, NTH, smemBC, stream>>>(E32, Yb, out);
}